// corr_loss_45578192945929
// MI455X (gfx1250) — compile-verified
//
#include <hip/hip_runtime.h>
#include <hip/hip_bf16.h>

#define NBINS 25
#define IMH 512
#define IMW 512
#define BATCH 32
#define PLANE (IMH * IMW)            // 262144 elements per (b,c) plane
#define NPLANES (BATCH * 3)          // 96
#define HSEG 8                       // histogram blocks per plane
#define MAIN_SPLIT 32                // main-kernel blocks per batch
#define MAIN_BLOCKS (BATCH * MAIN_SPLIT)  // 1024
#define TOTAL_N 25165824.0f          // 32*3*512*512

typedef __attribute__((ext_vector_type(2))) float v2f;
typedef __attribute__((ext_vector_type(8))) float v8f;

// ---------------------------------------------------------------------------
// Kernel 1: per-plane histogram of y (torch.histc(v, 25, 0, 1) semantics).
// 768 blocks, each covers 32768 elems of one plane with float4 loads.
// Per-wave privatized LDS bins -> u32 global atomics.
// ---------------------------------------------------------------------------
__global__ __launch_bounds__(256) void hist_kernel(const float* __restrict__ y,
                                                   unsigned int* __restrict__ ghist) {
    __shared__ unsigned int h[8 * 32];
    const int tid  = threadIdx.x;
    const int wave = tid >> 5;
    if (tid < 8 * 32) h[tid] = 0u;
    __syncthreads();

    const int plane = blockIdx.x / HSEG;
    const int seg   = blockIdx.x % HSEG;
    const float4* y4 = (const float4*)y;
    const size_t base4 = (size_t)plane * (PLANE / 4) + (size_t)seg * (PLANE / 4 / HSEG);

    for (int i = 0; i < 32; ++i) {
        float4 v = y4[base4 + (size_t)i * 256 + tid];
        float vv[4] = {v.x, v.y, v.z, v.w};
#pragma unroll
        for (int j = 0; j < 4; ++j) {
            float val = vv[j];
            int b = (int)(val * 25.0f);
            b = min(max(b, 0), NBINS - 1);
            if (val >= 0.0f && val <= 1.0f)
                atomicAdd(&h[wave * 32 + b], 1u);
        }
    }
    __syncthreads();
    if (tid < NBINS) {
        unsigned int s = 0;
        for (int wv = 0; wv < 8; ++wv) s += h[wv * 32 + tid];
        atomicAdd(&ghist[plane * NBINS + tid], s);
    }
}

// ---------------------------------------------------------------------------
// Kernel 2: Pearson correlations between channel histograms -> crg,crb,cgb.
// One wave32; lane b handles batch b.
// ---------------------------------------------------------------------------
__global__ void pearson_kernel(const unsigned int* __restrict__ ghist,
                               float* __restrict__ coeff) {
    int b = threadIdx.x;
    if (b >= BATCH) return;
    float h0[NBINS], h1[NBINS], h2[NBINS];
    float m0 = 0.f, m1 = 0.f, m2 = 0.f;
    for (int i = 0; i < NBINS; ++i) {
        h0[i] = (float)ghist[(b * 3 + 0) * NBINS + i];
        h1[i] = (float)ghist[(b * 3 + 1) * NBINS + i];
        h2[i] = (float)ghist[(b * 3 + 2) * NBINS + i];
        m0 += h0[i]; m1 += h1[i]; m2 += h2[i];
    }
    m0 /= NBINS; m1 /= NBINS; m2 /= NBINS;
    float c01 = 0.f, c02 = 0.f, c12 = 0.f, v0 = 0.f, v1 = 0.f, v2 = 0.f;
    for (int i = 0; i < NBINS; ++i) {
        float a = h0[i] - m0, g = h1[i] - m1, c = h2[i] - m2;
        c01 += a * g; c02 += a * c; c12 += g * c;
        v0 += a * a; v1 += g * g; v2 += c * c;
    }
    coeff[b * 3 + 0] = c01 / sqrtf(v0 * v1);  // crg
    coeff[b * 3 + 1] = c02 / sqrtf(v0 * v2);  // crb
    coeff[b * 3 + 2] = c12 / sqrtf(v1 * v2);  // cgb
}

// ---------------------------------------------------------------------------
// Kernel 3: main fused pass. Per 16-pixel group build B (4x16) directly in
// WMMA layout and compute Qd rows with V_WMMA_F32_16X16X4_F32.
//   B vgpr0: lanes0-15 = d_ch0(px), lanes16-31 = d_ch2(px)   (K=0 / K=2)
//   B vgpr1: lanes0-15 = d_ch1(px), lanes16-31 = 0           (K=1 / K=3)
//   A vgpr0: lanes0-2 = A[m][0], lanes16-18 = A[m][2], else 0
//   A vgpr1: lanes0-2 = A[m][1], else 0
// Rows >= 3 of A_pad are zero, so summing |D0|+|D1|+|D2| over all lanes is
// exactly sum|Qd| for the 16 pixels; |B0|+|B1| gives the L1 term.
// ---------------------------------------------------------------------------
__global__ __launch_bounds__(256) void main_kernel(const float* __restrict__ x,
                                                   const float* __restrict__ y,
                                                   const float* __restrict__ coeff,
                                                   float* __restrict__ partials) {
    const int b    = blockIdx.x / MAIN_SPLIT;
    const int r    = blockIdx.x % MAIN_SPLIT;
    const int tid  = threadIdx.x;
    const int lane = tid & 31;
    const int wave = tid >> 5;

    const int PXB = PLANE / MAIN_SPLIT;  // 8192 pixels per block
    const int PXW = PXB / 8;             // 1024 pixels per wave
    const int pxStart = r * PXB + wave * PXW;
    const size_t base = (size_t)b * 3 * PLANE;

    const float crg = coeff[b * 3 + 0];
    const float crb = coeff[b * 3 + 1];
    const float cgb = coeff[b * 3 + 2];

    // Build A_pad (16x4) in WMMA A-layout via lane-id selects.
    float a0v = 0.f, a1v = 0.f;
    if      (lane == 0)  { a0v = 1.f;  a1v = crg; }
    else if (lane == 1)  { a0v = crg;  a1v = 1.f; }
    else if (lane == 2)  { a0v = crb;  a1v = cgb; }
    else if (lane == 16) { a0v = crb; }
    else if (lane == 17) { a0v = cgb; }
    else if (lane == 18) { a0v = 1.f; }
    v2f amat; amat.x = a0v; amat.y = a1v;

    const bool   lowHalf = (lane < 16);
    const size_t chOff   = lowHalf ? (size_t)0 : (size_t)(2 * PLANE);
    const int    subpx   = lane & 15;

    float l1acc = 0.f, cracc = 0.f;
    v8f czero = {};

#pragma unroll 4
    for (int i = 0; i < PXW / 16; ++i) {
        const size_t px  = (size_t)(pxStart + i * 16 + subpx);
        const size_t aB0 = base + chOff + px;            // ch0 (low) / ch2 (high)
        const size_t aB1 = base + PLANE + px;            // ch1 (high-lane load aliases low's cachelines)
        const float b0 = x[aB0] - y[aB0];
        const float t1 = x[aB1] - y[aB1];
        const float b1 = lowHalf ? t1 : 0.f;

        l1acc += fabsf(b0) + fabsf(b1);

        v2f bmat; bmat.x = b0; bmat.y = b1;
        v8f dmat = __builtin_amdgcn_wmma_f32_16x16x4_f32(
            /*neg_a=*/false, amat, /*neg_b=*/false, bmat,
            /*c_mod=*/(short)0, czero, /*reuse_a=*/false, /*reuse_b=*/false);

        cracc += fabsf(dmat[0]) + fabsf(dmat[1]) + fabsf(dmat[2]);
    }

    // Deterministic block reduction.
    __shared__ float s1[256], s2[256];
    s1[tid] = l1acc; s2[tid] = cracc;
    __syncthreads();
    for (int s = 128; s > 0; s >>= 1) {
        if (tid < s) { s1[tid] += s1[tid + s]; s2[tid] += s2[tid + s]; }
        __syncthreads();
    }
    if (tid == 0) {
        partials[blockIdx.x * 2 + 0] = s1[0];
        partials[blockIdx.x * 2 + 1] = s2[0];
    }
}

// ---------------------------------------------------------------------------
// Kernel 4: deterministic final reduction of 1024 partial pairs -> loss.
// ---------------------------------------------------------------------------
__global__ __launch_bounds__(256) void finish_kernel(const float* __restrict__ partials,
                                                     float* __restrict__ out) {
    __shared__ float s1[256], s2[256];
    const int tid = threadIdx.x;
    float a = 0.f, c = 0.f;
    for (int k = tid; k < MAIN_BLOCKS; k += 256) {
        a += partials[k * 2 + 0];
        c += partials[k * 2 + 1];
    }
    s1[tid] = a; s2[tid] = c;
    __syncthreads();
    for (int s = 128; s > 0; s >>= 1) {
        if (tid < s) { s1[tid] += s1[tid + s]; s2[tid] += s2[tid + s]; }
        __syncthreads();
    }
    if (tid == 0) {
        out[0] = 0.8f * (s1[0] / TOTAL_N) + 0.2f * (s2[0] / TOTAL_N);
    }
}

// ---------------------------------------------------------------------------
extern "C" void kernel_launch(void* const* d_in, const int* in_sizes, int n_in,
                              void* d_out, int out_size, void* d_ws, size_t ws_size,
                              hipStream_t stream) {
    const float* x = (const float*)d_in[0];
    const float* y = (const float*)d_in[1];

    unsigned char* ws = (unsigned char*)d_ws;
    unsigned int* ghist    = (unsigned int*)ws;            // 96*25 u32  = 9600 B
    float*        coeff    = (float*)(ws + 9600);          // 96 floats  =  384 B
    float*        partials = (float*)(ws + 9984);          // 2048 floats = 8192 B

    hipMemsetAsync(ghist, 0, NPLANES * NBINS * sizeof(unsigned int), stream);
    hist_kernel<<<NPLANES * HSEG, 256, 0, stream>>>(y, ghist);
    pearson_kernel<<<1, 32, 0, stream>>>(ghist, coeff);
    main_kernel<<<MAIN_BLOCKS, 256, 0, stream>>>(x, y, coeff, partials);
    finish_kernel<<<1, 256, 0, stream>>>(partials, (float*)d_out);
}